// COMPASSNet_46325517255184
// MI455X (gfx1250) — compile-verified
//
#include <hip/hip_runtime.h>
#include <math.h>

// ---------------- problem constants (match reference) ----------------
#define D   32
#define H   4
#define P   529

#define TILE      16          // samples per WMMA tile (M dimension)
#define XSTRIDE   36          // padded LDS row stride (dwords) for the x tile
#define K4_WAVES  8           // waves per block in the compute kernel
#define K4_THREADS (K4_WAVES * 32)

// ---------------- CDNA5 feature detection ----------------
#if defined(__has_builtin)
#  if __has_builtin(__builtin_amdgcn_global_load_async_to_lds_b128)
#    define HAVE_ASYNC_LDS 1
#  endif
#  if __has_builtin(__builtin_amdgcn_s_wait_asynccnt)
#    define HAVE_WAIT_ASYNC 1
#  endif
#  if __has_builtin(__builtin_amdgcn_wmma_f32_16x16x4_f32)
#    define HAVE_WMMA_F32 1
#  endif
#endif

typedef float v2f __attribute__((ext_vector_type(2)));
typedef float v8f __attribute__((ext_vector_type(8)));
#if defined(HAVE_ASYNC_LDS)
typedef int v4i __attribute__((ext_vector_type(4)));
typedef __attribute__((address_space(1))) v4i* g_v4i_p;   // global
typedef __attribute__((address_space(3))) v4i* l_v4i_p;   // LDS
#endif

// ===================== K1: per-pattern histogram =====================
__global__ __launch_bounds__(256) void k1_hist(const int* __restrict__ pids,
                                               int* __restrict__ counts, int nB) {
    __shared__ int h[P];
    for (int i = threadIdx.x; i < P; i += blockDim.x) h[i] = 0;
    __syncthreads();
    const int b = blockIdx.x * blockDim.x + threadIdx.x;
    if (b < nB) atomicAdd(&h[pids[b]], 1);
    __syncthreads();
    for (int i = threadIdx.x; i < P; i += blockDim.x)
        if (h[i]) atomicAdd(&counts[i], h[i]);
}

// ===================== K2: exclusive scan (counts & tiles) =====================
__global__ __launch_bounds__(1024) void k2_scan(const int* __restrict__ counts,
                                                int* __restrict__ binStart,
                                                int* __restrict__ binCursor,
                                                int* __restrict__ tileOff,
                                                int* __restrict__ totT) {
    __shared__ int sc[1024];
    __shared__ int st[1024];
    const int i = threadIdx.x;
    const int c = (i < P) ? counts[i] : 0;
    const int t = (c + TILE - 1) >> 4;
    sc[i] = c; st[i] = t;
    __syncthreads();
    for (int off = 1; off < 1024; off <<= 1) {
        const int a  = (i >= off) ? sc[i - off] : 0;
        const int b2 = (i >= off) ? st[i - off] : 0;
        __syncthreads();
        sc[i] += a; st[i] += b2;
        __syncthreads();
    }
    if (i < P) {
        binStart[i]  = sc[i] - c;   // exclusive
        binCursor[i] = sc[i] - c;
        tileOff[i]   = st[i] - t;
    }
    if (i == 0) {
        binStart[P] = sc[1023];
        tileOff[P]  = st[1023];
        *totT       = st[1023];
    }
}

// ===================== K3: scatter sample indices into pattern bins =====================
__global__ __launch_bounds__(256) void k3_scatter(const int* __restrict__ pids,
                                                  int* __restrict__ binCursor,
                                                  int* __restrict__ sorted, int nB) {
    const int b = blockIdx.x * blockDim.x + threadIdx.x;
    if (b < nB) {
        const int pos = atomicAdd(&binCursor[pids[b]], 1);
        sorted[pos] = b;
    }
}

// ===================== K4: per-pattern tiles via V_WMMA_F32_16X16X4_F32 =====================
__global__ __launch_bounds__(K4_THREADS) void k4_compute(
    const float* __restrict__ x,      // (B, 32)
    const float* __restrict__ W1,     // (P, 32, 4)
    const float* __restrict__ b1,     // (P, 4)
    const float* __restrict__ W2,     // (P, 4)
    const float* __restrict__ b2,     // (P,)
    const int*   __restrict__ sorted,
    const int*   __restrict__ binStart,
    const int*   __restrict__ tileOff,
    const int*   __restrict__ totT,
    float*       __restrict__ out)
{
    __shared__ float tileLds[K4_WAVES][TILE * XSTRIDE];   // 2304 B / wave
    __shared__ float zLds[K4_WAVES][TILE];

    const int wave = threadIdx.x >> 5;
    const int lane = threadIdx.x & 31;
    const int t = blockIdx.x * K4_WAVES + wave;
    if (t >= *totT) return;                       // wave-uniform exit

    // --- find this tile's pattern (binary search over 530 tile offsets) ---
    int lo = 0, hi = P - 1;
    while (lo < hi) {
        const int mid = (lo + hi + 1) >> 1;
        if (tileOff[mid] <= t) lo = mid; else hi = mid - 1;
    }
    const int p     = lo;
    const int lt    = t - tileOff[p];
    const int start = binStart[p] + lt * TILE;
    const int cnt   = min(TILE, binStart[p + 1] - start);   // 1..16 valid rows

    float* tl = &tileLds[wave][0];
    const int m = lane & 15;          // M (sample) for A; N (column) for B/C/D
    const int g = lane >> 4;          // K-half selector

    // --- stage 16 x-rows into LDS: CDNA5 async-to-LDS, coalesced 128B rows ---
#pragma unroll
    for (int it = 0; it < (TILE * 8) / 32; ++it) {
        const int c   = lane + 32 * it;
        const int r   = c >> 3;                    // tile row
        const int cc  = c & 7;                     // 16B chunk in row
        const int row = min(r, cnt - 1);           // pad: duplicate last row
        const int sidx = sorted[start + row];
        const float4* src = (const float4*)x + (size_t)sidx * (D / 4) + cc;
        float* dst = tl + r * XSTRIDE + cc * 4;
#if defined(HAVE_ASYNC_LDS)
        __builtin_amdgcn_global_load_async_to_lds_b128((g_v4i_p)src, (l_v4i_p)dst, 0, 0);
#else
        *(float4*)dst = *src;
#endif
    }
#if defined(HAVE_WAIT_ASYNC)
    __builtin_amdgcn_s_wait_asynccnt(0);
#elif defined(HAVE_ASYNC_LDS)
    asm volatile("s_wait_asynccnt 0" ::: "memory");
#endif
    asm volatile("" ::: "memory");     // keep ds reads behind the wait

    // --- C init: b1 broadcast down the M dimension ---
    const float cinit = (m < H) ? b1[p * H + m] : 0.0f;
    v8f C;
#pragma unroll
    for (int v = 0; v < 8; ++v) C[v] = cinit;

    const float* __restrict__ wp = W1 + p * (D * H);

    // --- 8 chained wmma steps: (16x32) @ (32x4) in K=4 slabs ---
#pragma unroll
    for (int s = 0; s < 8; ++s) {
        const int k0 = 4 * s + 2 * g;
        // A: lane (m,g) holds x[m][k0], x[m][k0+1]  (NaN -> 0)
        v2f A = *(const v2f*)(tl + m * XSTRIDE + k0);
        A.x = (A.x == A.x) ? A.x : 0.0f;
        A.y = (A.y == A.y) ? A.y : 0.0f;
        // B: lane (n,g) holds W1[p][k0][n], W1[p][k0+1][n]; zero for n >= 4
        const float bx = wp[k0 * H + (m & 3)];
        const float by = wp[(k0 + 1) * H + (m & 3)];
        v2f Bv;
        Bv.x = (m < H) ? bx : 0.0f;
        Bv.y = (m < H) ? by : 0.0f;
#if defined(HAVE_WMMA_F32)
        C = __builtin_amdgcn_wmma_f32_16x16x4_f32(
            /*neg_a=*/false, A, /*neg_b=*/false, Bv,
            /*c_mod=*/(short)0, C, /*reuse_a=*/false, /*reuse_b=*/false);
#else
        // scalar emulation of the same D layout (M=v+8g, N=m)
#pragma unroll
        for (int v = 0; v < 8; ++v) {
            const int M = v + 8 * g;
#pragma unroll
            for (int kk = 0; kk < 4; ++kk) {
                float xa = tl[M * XSTRIDE + 4 * s + kk];
                xa = (xa == xa) ? xa : 0.0f;
                const float wb = (m < H) ? wp[(4 * s + kk) * H + (m & 3)] : 0.0f;
                C[v] = fmaf(xa, wb, C[v]);
            }
        }
#endif
    }

    // --- epilogue: z[M] = sum_n tanh(D[M][n]) * W2[n]  (n = lane&15) ---
    const float w2n = (m < H) ? W2[p * H + m] : 0.0f;
#pragma unroll
    for (int v = 0; v < 8; ++v) {
        float tv = tanhf(C[v]) * w2n;       // lanes n>=4 contribute exactly 0
        tv += __shfl_xor(tv, 1, 32);
        tv += __shfl_xor(tv, 2, 32);        // lanes n in {0..3} hold full sum
        if (m == 0) zLds[wave][v + 8 * g] = tv;
    }

    if (lane < TILE) {
        const float z = zLds[wave][lane] + b2[p];
        const float y = 1.0f / (1.0f + expf(-z));
        if (lane < cnt) out[sorted[start + lane]] = y;
    }
}

// ===================== host side =====================
extern "C" void kernel_launch(void* const* d_in, const int* in_sizes, int n_in,
                              void* d_out, int out_size, void* d_ws, size_t ws_size,
                              hipStream_t stream) {
    const float* x   = (const float*)d_in[0];
    const int*   pid = (const int*)d_in[1];
    const float* W1  = (const float*)d_in[2];
    const float* b1  = (const float*)d_in[3];
    const float* W2  = (const float*)d_in[4];
    const float* b2  = (const float*)d_in[5];
    float* out = (float*)d_out;
    const int nB = in_sizes[1];                    // pattern_ids has B elements

    // workspace layout (ints)
    int* w         = (int*)d_ws;
    int* counts    = w;                            // P
    int* binStart  = counts + P;                   // P+1
    int* binCursor = binStart + (P + 1);           // P
    int* tileOff   = binCursor + P;                // P+1
    int* totT      = tileOff + (P + 1);            // 1
    int* sorted    = totT + 1;                     // nB

    (void)hipMemsetAsync(counts, 0, P * sizeof(int), stream);

    const int blocks = (nB + 255) / 256;
    k1_hist<<<blocks, 256, 0, stream>>>(pid, counts, nB);
    k2_scan<<<1, 1024, 0, stream>>>(counts, binStart, binCursor, tileOff, totT);
    k3_scatter<<<blocks, 256, 0, stream>>>(pid, binCursor, sorted, nB);

    // static upper bound on tile count (actual T read on-device)
    const int maxT = (nB + TILE - 1) / TILE + P;
    k4_compute<<<(maxT + K4_WAVES - 1) / K4_WAVES, K4_THREADS, 0, stream>>>(
        x, W1, b1, W2, b2, sorted, binStart, tileOff, totT, out);
}